// MultitaskLossA_1743756722577
// MI455X (gfx1250) — compile-verified
//
#include <hip/hip_runtime.h>
#include <hip/hip_bf16.h>
#include <math.h>

typedef __attribute__((ext_vector_type(2))) float v2f;
typedef __attribute__((ext_vector_type(8))) float v8f;

#define T_DIM 8
#define S_DIM 16
#define BLOCK 256

__device__ __forceinline__ float bce_term(float x, float y) {
    // max(x,0) - x*y + log1p(exp(-|x|))  (numerically stable BCE-with-logits)
    return fmaxf(x, 0.0f) - x * y + log1pf(__expf(-fabsf(x)));
}

__global__ __launch_bounds__(BLOCK) void multitask_loss_partial(
    const float* __restrict__ ypb,   // [B]      binary logits
    const float* __restrict__ ypt,   // [B,8]    type logits
    const float* __restrict__ yps,   // [B,16]   source logits
    const float* __restrict__ ytb,   // [B]      binary targets
    const float* __restrict__ ytt,   // [B,8]    type targets
    const int*   __restrict__ yts,   // [B]      source class index
    float* __restrict__ partials,    // [gridDim.x]
    int B, float invB, float invBT)
{
    float acc = 0.0f;
    const int stride = gridDim.x * blockDim.x;
    for (int row = (int)(blockIdx.x * blockDim.x + threadIdx.x); row < B; row += stride) {
        // Prefetch next grid-stride row (lowers to global_prefetch_b8 on gfx1250).
        int nrow = row + stride;
        if (nrow < B) {
            __builtin_prefetch(yps + (size_t)nrow * S_DIM, 0, 0);
            __builtin_prefetch(ypt + (size_t)nrow * T_DIM, 0, 0);
        }

        // ---- binary BCE ----
        float xb = ypb[row];
        float yb = ytb[row];
        float bin = bce_term(xb, yb);

        // ---- type BCE (8 wide, b128 loads) ----
        const float4* tp = (const float4*)(ypt + (size_t)row * T_DIM);
        const float4* tt = (const float4*)(ytt + (size_t)row * T_DIM);
        float4 tp0 = tp[0], tp1 = tp[1];
        float4 tt0 = tt[0], tt1 = tt[1];
        float typeSum =
            bce_term(tp0.x, tt0.x) + bce_term(tp0.y, tt0.y) +
            bce_term(tp0.z, tt0.z) + bce_term(tp0.w, tt0.w) +
            bce_term(tp1.x, tt1.x) + bce_term(tp1.y, tt1.y) +
            bce_term(tp1.z, tt1.z) + bce_term(tp1.w, tt1.w);

        // ---- source cross-entropy over 16 classes (4x b128 loads) ----
        const float4* sp = (const float4*)(yps + (size_t)row * S_DIM);
        float4 s0 = sp[0], s1 = sp[1], s2 = sp[2], s3 = sp[3];
        float v[S_DIM] = { s0.x, s0.y, s0.z, s0.w,
                           s1.x, s1.y, s1.z, s1.w,
                           s2.x, s2.y, s2.z, s2.w,
                           s3.x, s3.y, s3.z, s3.w };
        float m = v[0];
        int amax = 0;                      // first-occurrence argmax (jnp semantics)
        #pragma unroll
        for (int i = 1; i < S_DIM; ++i) {
            if (v[i] > m) { m = v[i]; amax = i; }
        }
        float se = 0.0f;
        #pragma unroll
        for (int i = 0; i < S_DIM; ++i) se += __expf(v[i] - m);
        int tgt = yts[row];
        float nll = (m + __logf(se)) - v[tgt];

        // ---- consistency indicators (non-differentiable counts) ----
        bool pb = xb    > 0.0f;            // round(sigmoid(x)) == 1  <=>  x > 0
        bool pt = tp0.x > 0.0f;            // type column 0
        bool srcInc = pb ? (amax == 0) : (amax > 0);
        bool typInc = pb ? (!pt)       : pt;
        float cons = (srcInc ? 1.0f : 0.0f) + (typInc ? 1.0f : 0.0f);

        // All terms pre-scaled so a single float reduction yields the final loss.
        acc += (bin + nll + cons) * invB + typeSum * invBT;
    }

    // Block tree reduction in LDS (deterministic order).
    __shared__ float sdata[BLOCK];
    sdata[threadIdx.x] = acc;
    __syncthreads();
    #pragma unroll
    for (int s = BLOCK / 2; s > 0; s >>= 1) {
        if ((int)threadIdx.x < s) sdata[threadIdx.x] += sdata[threadIdx.x + s];
        __syncthreads();
    }
    if (threadIdx.x == 0) partials[blockIdx.x] = sdata[0];
}

// Final reduction: one wave (32 lanes, EXEC all-1s as WMMA requires).
// Cross-lane sum via V_WMMA_F32_16X16X4_F32 with B = ones:
//   A[m][0] = acc(lane m), A[m][2] = acc(lane m+16), rest 0
//   => D[m][n] = acc_m + acc_{m+16} for every n.
// Summing a lane's 8 C/D VGPRs gives rows 0..7 in lanes 0..15 and rows 8..15
// in lanes 16..31; total = lane0 + lane16.
__global__ __launch_bounds__(32) void final_reduce_wmma(
    const float* __restrict__ partials, int n, float* __restrict__ out)
{
    const int lane = threadIdx.x;
    float acc = 0.0f;
    for (int i = lane; i < n; i += 32) acc += partials[i];

    v2f a; a.x = acc;  a.y = 0.0f;
    v2f b; b.x = 1.0f; b.y = 1.0f;
    v8f c = {};
    c = __builtin_amdgcn_wmma_f32_16x16x4_f32(
        /*neg_a=*/false, a, /*neg_b=*/false, b,
        /*c_mod=*/(short)0, c, /*reuse_a=*/false, /*reuse_b=*/false);

    float s = c[0] + c[1] + c[2] + c[3] + c[4] + c[5] + c[6] + c[7];
    float total = __shfl(s, 0, 32) + __shfl(s, 16, 32);
    if (lane == 0) out[0] = total;
}

extern "C" void kernel_launch(void* const* d_in, const int* in_sizes, int n_in,
                              void* d_out, int out_size, void* d_ws, size_t ws_size,
                              hipStream_t stream) {
    const float* ypb = (const float*)d_in[0];
    const float* ypt = (const float*)d_in[1];
    const float* yps = (const float*)d_in[2];
    const float* ytb = (const float*)d_in[3];
    const float* ytt = (const float*)d_in[4];
    const int*   yts = (const int*)  d_in[5];

    const int B = in_sizes[0];
    const int T = (B > 0) ? (in_sizes[1] / B) : T_DIM;   // == 8 per reference

    float* partials = (float*)d_ws;
    int blocks = 4096;                                   // streaming parallelism for 280 MB
    int maxBlocks = (int)(ws_size / sizeof(float));
    if (blocks > maxBlocks) blocks = maxBlocks;
    if (blocks < 1) blocks = 1;

    const float invB  = 1.0f / (float)B;
    const float invBT = 1.0f / ((float)B * (float)T);

    multitask_loss_partial<<<blocks, BLOCK, 0, stream>>>(
        ypb, ypt, yps, ytb, ytt, yts, partials, B, invB, invBT);
    final_reduce_wmma<<<1, 32, 0, stream>>>(partials, blocks, (float*)d_out);
}